// MultiClassDiceScorewithBG_28656021799145
// MI455X (gfx1250) — compile-verified
//
#include <hip/hip_runtime.h>
#include <stdint.h>

typedef unsigned int u32;
typedef __attribute__((ext_vector_type(4))) int v4i;

#define AS1 __attribute__((address_space(1)))
#define AS3 __attribute__((address_space(3)))

#define NCLS            5
#define TILES_PER_BLOCK 16
#define TILE_VOX        1024   // 256 threads * 4 voxels
#define DEPTH           4      // async pipeline depth (tiles in flight)

// ---------------- CDNA5 async-to-LDS feature detection -----------------
#if defined(__HIP_DEVICE_COMPILE__) && defined(__gfx1250__) && \
    __has_builtin(__builtin_amdgcn_global_load_async_to_lds_b128)
#define USE_ASYNC 1
#else
#define USE_ASYNC 0
#endif

__device__ __forceinline__ void wait_async_n(int n) {
#if USE_ASYNC
#if __has_builtin(__builtin_amdgcn_s_wait_asynccnt)
  switch (n) {
    case 0:  __builtin_amdgcn_s_wait_asynccnt(0); break;
    case 2:  __builtin_amdgcn_s_wait_asynccnt(2); break;
    case 4:  __builtin_amdgcn_s_wait_asynccnt(4); break;
    default: __builtin_amdgcn_s_wait_asynccnt(6); break;
  }
#else
  switch (n) {
    case 0:  asm volatile("s_wait_asynccnt 0" ::: "memory"); break;
    case 2:  asm volatile("s_wait_asynccnt 2" ::: "memory"); break;
    case 4:  asm volatile("s_wait_asynccnt 4" ::: "memory"); break;
    default: asm volatile("s_wait_asynccnt 6" ::: "memory"); break;
  }
#endif
  asm volatile("" ::: "memory");  // keep LDS reads below the wait
#else
  (void)n;
#endif
}

// Packed per-lane counting: 5 classes x 6-bit fields in one u32 per array.
// One increment = shift + add; one instruction covers 32 voxels wave-wide.
__device__ __forceinline__ void acc4(const float4& p4, const int4& m4,
                                     u32& pd1, u32& pd2, u32& pnm) {
  const float pf[4] = {p4.x, p4.y, p4.z, p4.w};
  const int   mm[4] = {m4.x, m4.y, m4.z, m4.w};
#pragma unroll
  for (int j = 0; j < 4; ++j) {
    const int pi = (int)pf[j];          // labels are exact small floats
    const int mi = mm[j];
    const u32 tp = 1u << (pi * 6);
    const u32 tm = 1u << (mi * 6);
    pd1 += tp;
    pd2 += tm;
    pnm += (pi == mi) ? tp : 0u;        // pi==mi implies same field
  }
}

__device__ __forceinline__ void flush_packed(u32& pd1, u32& pd2, u32& pnm,
                                             u32 d1[NCLS], u32 d2[NCLS],
                                             u32 nm[NCLS]) {
#pragma unroll
  for (int c = 0; c < NCLS; ++c) {
    d1[c] += (pd1 >> (6 * c)) & 63u;
    d2[c] += (pd2 >> (6 * c)) & 63u;
    nm[c] += (pnm >> (6 * c)) & 63u;
  }
  pd1 = pd2 = pnm = 0u;
}

__global__ __launch_bounds__(256) void dice_hist_kernel(
    const float* __restrict__ outp, const int* __restrict__ mask,
    u32* __restrict__ gcnt) {
  const int tid  = threadIdx.x;
  const int lane = tid & 31;
  const int wave = tid >> 5;

  u32 d1[NCLS] = {0, 0, 0, 0, 0};
  u32 d2[NCLS] = {0, 0, 0, 0, 0};
  u32 nm[NCLS] = {0, 0, 0, 0, 0};
  u32 pd1 = 0, pd2 = 0, pnm = 0;

  // float4/int4 index of this thread's first tile element
  const size_t base4 = (size_t)blockIdx.x * (TILES_PER_BLOCK * 256) +
                       (size_t)wave * 32 + (size_t)lane;
  const float4* __restrict__ g_out = (const float4*)outp;
  const int4*   __restrict__ g_msk = (const int4*)mask;

#if USE_ASYNC
  // Per-wave private staging: 32 lanes * 16B * {out,mask} * DEPTH buffers.
  __shared__ __align__(16) unsigned char stage[8][DEPTH][2][512];

#define ISSUE_TILE(i, b)                                                       \
  do {                                                                         \
    __builtin_amdgcn_global_load_async_to_lds_b128(                            \
        (AS1 v4i*)(g_out + base4 + (size_t)(i) * 256),                         \
        (AS3 v4i*)&stage[wave][(b)][0][lane * 16], 0, 0);                      \
    __builtin_amdgcn_global_load_async_to_lds_b128(                            \
        (AS1 v4i*)(g_msk + base4 + (size_t)(i) * 256),                         \
        (AS3 v4i*)&stage[wave][(b)][1][lane * 16], 0, 0);                      \
  } while (0)

#pragma unroll
  for (int i = 0; i < DEPTH; ++i) ISSUE_TILE(i, i);

#pragma unroll
  for (int i = 0; i < TILES_PER_BLOCK; ++i) {
    const int remaining      = TILES_PER_BLOCK - 1 - i;
    const int inflight_after = remaining < (DEPTH - 1) ? remaining : (DEPTH - 1);
    wait_async_n(2 * inflight_after);  // async loads complete in order

    const int b = i & (DEPTH - 1);
    const float4 p4 = *(const float4*)&stage[wave][b][0][lane * 16];
    const int4   m4 = *(const int4*)&stage[wave][b][1][lane * 16];
    acc4(p4, m4, pd1, pd2, pnm);

    if (i + DEPTH < TILES_PER_BLOCK) ISSUE_TILE(i + DEPTH, b);
    if (i == 7) flush_packed(pd1, pd2, pnm, d1, d2, nm);  // 6-bit cap: 63
  }
#undef ISSUE_TILE
#else
#pragma unroll
  for (int i = 0; i < TILES_PER_BLOCK; ++i) {
    const float4 p4 = g_out[base4 + (size_t)i * 256];
    const int4   m4 = g_msk[base4 + (size_t)i * 256];
    acc4(p4, m4, pd1, pd2, pnm);
    if (i == 7) flush_packed(pd1, pd2, pnm, d1, d2, nm);
  }
#endif
  flush_packed(pd1, pd2, pnm, d1, d2, nm);

  // Wave32 butterfly reduction of the 15 counters.
#pragma unroll
  for (int off = 16; off > 0; off >>= 1) {
#pragma unroll
    for (int c = 0; c < NCLS; ++c) {
      d1[c] += __shfl_xor(d1[c], off, 32);
      d2[c] += __shfl_xor(d2[c], off, 32);
      nm[c] += __shfl_xor(nm[c], off, 32);
    }
  }

  __shared__ u32 shc[3 * NCLS];
  if (tid < 3 * NCLS) shc[tid] = 0u;
  __syncthreads();
  if (lane == 0) {
#pragma unroll
    for (int c = 0; c < NCLS; ++c) {
      atomicAdd(&shc[c], d1[c]);
      atomicAdd(&shc[NCLS + c], d2[c]);
      atomicAdd(&shc[2 * NCLS + c], nm[c]);
    }
  }
  __syncthreads();
  if (tid < 3 * NCLS) atomicAdd(&gcnt[tid], shc[tid]);
}

// Generic scalar tail (not used for the 128x512x512 shape, which tiles exactly)
__global__ void dice_tail_kernel(const float* __restrict__ outp,
                                 const int* __restrict__ mask,
                                 u32* __restrict__ gcnt, long long start,
                                 long long n) {
  u32 d1[NCLS] = {0, 0, 0, 0, 0};
  u32 d2[NCLS] = {0, 0, 0, 0, 0};
  u32 nm[NCLS] = {0, 0, 0, 0, 0};
  for (long long i = start + threadIdx.x; i < n; i += blockDim.x) {
    const int pi = (int)outp[i];
    const int mi = mask[i];
#pragma unroll
    for (int c = 0; c < NCLS; ++c) {
      d1[c] += (pi == c);
      d2[c] += (mi == c);
      nm[c] += (pi == c) & (pi == mi);
    }
  }
#pragma unroll
  for (int c = 0; c < NCLS; ++c) {
    if (d1[c]) atomicAdd(&gcnt[c], d1[c]);
    if (d2[c]) atomicAdd(&gcnt[NCLS + c], d2[c]);
    if (nm[c]) atomicAdd(&gcnt[2 * NCLS + c], nm[c]);
  }
}

__global__ void dice_zero_kernel(u32* __restrict__ gcnt) {
  if (threadIdx.x < 3 * NCLS) gcnt[threadIdx.x] = 0u;
}

__global__ void dice_finalize_kernel(const u32* __restrict__ gcnt,
                                     float* __restrict__ out) {
  if (threadIdx.x == 0 && blockIdx.x == 0) {
    float s = 0.0f;
#pragma unroll
    for (int c = 0; c < NCLS; ++c) {
      const float num = (float)gcnt[2 * NCLS + c];
      const float den = (float)gcnt[c] + (float)gcnt[NCLS + c];
      s += 2.0f * ((num + 1e-7f) / (den + 1e-7f));
    }
    out[0] = s / (float)NCLS;
  }
}

extern "C" void kernel_launch(void* const* d_in, const int* in_sizes, int n_in,
                              void* d_out, int out_size, void* d_ws,
                              size_t ws_size, hipStream_t stream) {
  (void)n_in; (void)out_size; (void)ws_size;
  const float* outp = (const float*)d_in[0];
  const int*   mask = (const int*)d_in[1];
  u32*   cnt = (u32*)d_ws;
  float* out = (float*)d_out;

  const long long n      = (long long)in_sizes[0];  // 33,554,432
  const long long tiles  = n / TILE_VOX;
  const long long blocks = tiles / TILES_PER_BLOCK; // 2048 for this shape
  const long long nMain  = blocks * (long long)TILES_PER_BLOCK * TILE_VOX;

  dice_zero_kernel<<<1, 32, 0, stream>>>(cnt);
  if (blocks > 0)
    dice_hist_kernel<<<dim3((unsigned)blocks), dim3(256), 0, stream>>>(outp, mask, cnt);
  if (nMain < n)
    dice_tail_kernel<<<1, 256, 0, stream>>>(outp, mask, cnt, nMain, n);
  dice_finalize_kernel<<<1, 32, 0, stream>>>(cnt, out);
}